// GATLayer_65283502899798
// MI455X (gfx1250) — compile-verified
//
#include <hip/hip_runtime.h>
#include <math.h>

typedef __attribute__((ext_vector_type(2))) float v2f;
typedef __attribute__((ext_vector_type(8))) float v8f;

#define D 128
#define NEG_SLOPE 0.01f

// ---------------------------------------------------------------------------
// Kernel 1: z = h @ W^T  via V_WMMA_F32_16X16X4_F32.
// Block = 256 threads (8 waves). Block b computes rows [16b, 16b+16) of z.
// Wave w computes the 16x16 tile at columns [16w, 16w+16).
// W (128x128 fp32, 64KB) is staged in LDS once per block.
// A layout (16x4 f32): lane&15 = M row, lane>>4 selects K pair, 2 floats/lane.
// B layout (4x16 f32): lane&15 = N col, lane>>4 selects K pair; element
//   B[k][n] = W[n][k], i.e. a contiguous read of W row n. Both are float2 loads.
// C/D layout: VGPR v holds (M = v + 8*(lane>>4), N = lane&15).
// ---------------------------------------------------------------------------
__global__ void __launch_bounds__(256)
gat_gemm_wmma(const float* __restrict__ h, const float* __restrict__ W,
              float* __restrict__ z, int N) {
  __shared__ float lw[D * D];  // 64 KB, fits in 320KB WGP LDS

  {
    const float4* srcW = (const float4*)W;
    float4* dstW = (float4*)lw;
    for (int i = threadIdx.x; i < (D * D) / 4; i += blockDim.x) dstW[i] = srcW[i];
  }
  __syncthreads();

  const int wave = threadIdx.x >> 5;   // 0..7 -> column tile
  const int lane = threadIdx.x & 31;
  const int half = lane >> 4;          // selects K pair within 4-wide K step
  const int l15  = lane & 15;

  const int m0 = blockIdx.x * 16;
  const int n0 = wave * 16;

  const int arow = (m0 + l15 < N) ? (m0 + l15) : (N - 1);  // clamp tail reads
  const float* aptr = h + (size_t)arow * D + 2 * half;
  const float* bptr = lw + (n0 + l15) * D + 2 * half;

  v8f c = {0.f, 0.f, 0.f, 0.f, 0.f, 0.f, 0.f, 0.f};

#pragma unroll
  for (int k = 0; k < D; k += 4) {
    v2f a = *(const v2f*)(aptr + k);
    v2f b = *(const v2f*)(bptr + k);
    // (neg_a, A, neg_b, B, c_mod, C, reuse_a, reuse_b)
    c = __builtin_amdgcn_wmma_f32_16x16x4_f32(false, a, false, b,
                                              (short)0, c, false, false);
  }

#pragma unroll
  for (int v = 0; v < 8; ++v) {
    int M = m0 + v + 8 * half;
    if (M < N) z[(size_t)M * D + n0 + l15] = c[v];
  }
}

// ---------------------------------------------------------------------------
// Kernel 2: logits_l[n] = z[n,:] . a_l ; logits_r[n] = z[n,:] . a_r
// One wave (32 lanes x float4 = 128) per node; wave32 shuffle reduction.
// ---------------------------------------------------------------------------
__global__ void __launch_bounds__(256)
gat_logits(const float* __restrict__ z, const float* __restrict__ attn_w,
           float* __restrict__ ll, float* __restrict__ lr, int N) {
  const int node = blockIdx.x * (blockDim.x >> 5) + (threadIdx.x >> 5);
  const int lane = threadIdx.x & 31;
  if (node >= N) return;

  float4 zv = ((const float4*)(z + (size_t)node * D))[lane];
  float4 av = ((const float4*)(attn_w))[lane];          // a_l = attn_w[0:128]
  float4 bv = ((const float4*)(attn_w + D))[lane];      // a_r = attn_w[128:256]

  float sl = zv.x * av.x + zv.y * av.y + zv.z * av.z + zv.w * av.w;
  float sr = zv.x * bv.x + zv.y * bv.y + zv.z * bv.z + zv.w * bv.w;

#pragma unroll
  for (int off = 16; off > 0; off >>= 1) {
    sl += __shfl_xor(sl, off, 32);
    sr += __shfl_xor(sr, off, 32);
  }
  if (lane == 0) { ll[node] = sl; lr[node] = sr; }
}

// ---------------------------------------------------------------------------
// Kernel 3: e[i] = leaky_relu(ll[src[i]] + lr[dst[i]])
// ---------------------------------------------------------------------------
__global__ void __launch_bounds__(256)
gat_edge_logits(const int* __restrict__ src, const int* __restrict__ dst,
                const float* __restrict__ ll, const float* __restrict__ lr,
                float* __restrict__ e, int E) {
  int i = blockIdx.x * blockDim.x + threadIdx.x;
  if (i >= E) return;
  float x = ll[src[i]] + lr[dst[i]];
  e[i] = (x > 0.f) ? x : NEG_SLOPE * x;
}

// ---------------------------------------------------------------------------
// Kernel 4: node-centric softmax + aggregation. dst is sorted, so each node's
// edges are a contiguous range found by binary search. One wave per node:
//   pass 1: segment max (shuffle reduce)
//   pass 2: sum of exp(e - m) (shuffle reduce)
//   pass 3: out[n,:] += (exp/s) * z[src]  -- 32 lanes x float4 = 128 features,
//           coalesced 512B row gathers (the bandwidth-dominant step).
// Deterministic: no atomics; out written exactly once per node.
// ---------------------------------------------------------------------------
__global__ void __launch_bounds__(256)
gat_aggregate(const int* __restrict__ src, const int* __restrict__ dst,
              const float* __restrict__ e, const float* __restrict__ z,
              float* __restrict__ out, int N, int E) {
  const int node = blockIdx.x * (blockDim.x >> 5) + (threadIdx.x >> 5);
  const int lane = threadIdx.x & 31;
  if (node >= N) return;

  // lower bound
  int lo = 0, hi = E;
  while (lo < hi) { int mid = (lo + hi) >> 1; if (dst[mid] < node) lo = mid + 1; else hi = mid; }
  const int beg = lo;
  // upper bound
  hi = E;
  while (lo < hi) { int mid = (lo + hi) >> 1; if (dst[mid] <= node) lo = mid + 1; else hi = mid; }
  const int end = lo;

  float4 acc = {0.f, 0.f, 0.f, 0.f};

  if (beg < end) {
    // segment max
    float m = -INFINITY;
    for (int i = beg + lane; i < end; i += 32) m = fmaxf(m, e[i]);
#pragma unroll
    for (int off = 16; off > 0; off >>= 1) m = fmaxf(m, __shfl_xor(m, off, 32));
    // m is finite here (edges exist), matching reference's where(isfinite,...)

    // segment sum of exp
    float s = 0.f;
    for (int i = beg + lane; i < end; i += 32) s += expf(e[i] - m);
#pragma unroll
    for (int off = 16; off > 0; off >>= 1) s += __shfl_xor(s, off, 32);
    const float inv_s = 1.f / s;

    // weighted feature aggregation
    for (int i = beg; i < end; ++i) {
      const float w = expf(e[i] - m) * inv_s;     // uniform across wave
      const int sn = src[i];                      // broadcast scalar load
      float4 zv = ((const float4*)(z + (size_t)sn * D))[lane];
      acc.x += w * zv.x; acc.y += w * zv.y; acc.z += w * zv.z; acc.w += w * zv.w;
    }
  }
  // always store (d_out is poisoned; isolated nodes must output zeros)
  ((float4*)(out + (size_t)node * D))[lane] = acc;
}

// ---------------------------------------------------------------------------
extern "C" void kernel_launch(void* const* d_in, const int* in_sizes, int n_in,
                              void* d_out, int out_size, void* d_ws, size_t ws_size,
                              hipStream_t stream) {
  const float* h      = (const float*)d_in[0];   // (N, 128) fp32
  const int*   src    = (const int*)d_in[1];     // (E,) i32
  const int*   dst    = (const int*)d_in[2];     // (E,) i32 (sorted)
  const float* W_fc   = (const float*)d_in[3];   // (128, 128) fp32
  const float* attn_w = (const float*)d_in[4];   // (1, 256) fp32

  const int N = in_sizes[0] / D;
  const int E = in_sizes[1];

  float* ws = (float*)d_ws;
  float* z  = ws;                          // N*128 floats
  float* ll = z + (size_t)N * D;           // N floats
  float* lr = ll + N;                      // N floats
  float* e  = lr + N;                      // E floats

  float* out = (float*)d_out;

  const int mtiles = (N + 15) / 16;
  gat_gemm_wmma<<<mtiles, 256, 0, stream>>>(h, W_fc, z, N);

  const int wpb = 256 / 32;  // 8 waves (nodes) per block
  gat_logits<<<(N + wpb - 1) / wpb, 256, 0, stream>>>(z, attn_w, ll, lr, N);

  gat_edge_logits<<<(E + 255) / 256, 256, 0, stream>>>(src, dst, ll, lr, e, E);

  gat_aggregate<<<(N + wpb - 1) / wpb, 256, 0, stream>>>(src, dst, e, z, out, N, E);
}